// RoPE_78340203479350
// MI455X (gfx1250) — compile-verified
//
#include <hip/hip_runtime.h>
#include <hip/hip_bf16.h>
#include <stdint.h>

// RoPE (B=4,H=32,S=8192,D=64, f32) for MI455X / gfx1250.
//
// Roofline: 537 MB streamed @ 23.3 TB/s => ~23 us floor; memory-bound.
// Strategy:
//   * TDM (tensor_load_to_lds) DMAs a 2D tile (16 rows x 512 contiguous
//     floats, row stride S*D) into LDS asynchronously (TENSORcnt).
//   * While the DMA runs, the block builds a 256-entry cos/sin table in LDS
//     (angles depend only on (s, j); 16 rows share it -> sincos work /16,
//     globally /16 too, as each row-block rebuilds it).
//   * Compute: ds_load_b128 tile + table (table read is loop-invariant and
//     hoisted to a single ds load), 8 VALU fmas, non-temporal
//     global_store_b128 straight to the output (streaming data, no reuse ->
//     don't churn the 192MB L2 with write-allocate).
//   * 34 KB LDS/block -> ~9 blocks resident per 320KB WGP; their async
//     TDM loads overlap other blocks' compute.

typedef unsigned int v4u __attribute__((ext_vector_type(4)));
typedef int          v8i __attribute__((ext_vector_type(8)));
typedef int          v4i __attribute__((ext_vector_type(4)));
typedef float        v4f __attribute__((ext_vector_type(4)));   // native vector (nontemporal-store friendly)

#define TILE_S 8     // positions per block
#define TILE_R 16    // (b,h) rows per block
#define DHEAD  64    // head dim (reference constant)
#define ROW_F  (TILE_S * DHEAD)          // 512 floats, contiguous in memory
#define TILE_F (TILE_R * ROW_F)          // 8192 floats = 32 KB
#define TILE_V4 (TILE_F / 4)             // 2048 float4
#define LOG2_THETA_OVER_HALF_D 0.4152410118609203f   // log2(10000)/32

__global__ __launch_bounds__(256)
void rope_tdm_kernel(const float* __restrict__ x,
                     const int*   __restrict__ tp,
                     float* __restrict__ out,
                     int S) {
    __shared__ v4f tile4[TILE_V4];    // TDM destination
    __shared__ v4f table4[TILE_S * (DHEAD / 4)];  // (c0,s0,c1,s1) per pair-quad

    const int tid = threadIdx.x;
    const int s0  = blockIdx.x * TILE_S;     // first position of tile
    const int r0  = blockIdx.y * TILE_R;     // first (b*H+h) row of tile

    const size_t rowStride = (size_t)S * DHEAD;               // floats
    const size_t gbase     = (size_t)r0 * rowStride + (size_t)s0 * DHEAD; // floats

    // ---- Issue TDM 2D tile load: dim0 = 512 contiguous floats,
    //      dim1 = 16 rows, tensor_dim0_stride = S*D. (one wave issues;
    //      TDM ignores EXEC, descriptor is uniform -> lives in SGPRs) ----
    if (tid < 32) {
        const unsigned lds_off = (unsigned)(uintptr_t)(void*)tile4; // LDS byte offset (flat addr low 32 bits)
        const uint64_t gaddr   = (uint64_t)(uintptr_t)(x + gbase);  // global byte address of tile start

        v4u g0;
        g0[0] = 1u;                                 // count=1 (valid), user mode
        g0[1] = lds_off;                            // lds_addr [63:32]
        g0[2] = (unsigned)(gaddr & 0xFFFFFFFFu);    // global_addr [95:64]
        g0[3] = (unsigned)((gaddr >> 32) & 0x1FFFFFFu) | (2u << 30); // addr[56:32] | type=2

        v8i g1;
        g1[0] = (int)(2u << 16);                    // data_size=2 (4B); wg_mask=0; no pad/iter
        g1[1] = (int)((unsigned)(ROW_F & 0xFFFF) << 16);   // tensor_dim0 low16 (512)
        g1[2] = (int)(((unsigned)ROW_F >> 16) | ((unsigned)TILE_R << 16)); // dim0 hi16 | tensor_dim1 lo16
        g1[3] = (int)((unsigned)ROW_F << 16);       // tensor_dim1 hi16 (0) | tile_dim0 (512)
        g1[4] = (int)TILE_R;                        // tile_dim1 (16) | tile_dim2 (0)
        g1[5] = (int)(unsigned)rowStride;           // tensor_dim0_stride low32 (S*D)
        g1[6] = 0;                                  // dim0_stride hi16 | dim1_stride lo16
        g1[7] = 0;

        v4i z4 = {0, 0, 0, 0};
        v8i z8 = {0, 0, 0, 0, 0, 0, 0, 0};
        // clang-23 / therock-10.0 signature: (g0, g1, g2, g3, g4, cpol)
        __builtin_amdgcn_tensor_load_to_lds(g0, g1, z4, z4, z8, 0); // TENSORcnt++
    }

    // ---- Build cos/sin table while the DMA is in flight.
    //      tid -> (pos p = tid>>5 in [0,8), j = tid&31 in [0,32)).
    {
        const int p = tid >> 5;
        const int j = tid & 31;
        const float pos = (float)tp[s0 + p];
        const float inv = exp2f(-LOG2_THETA_OVER_HALF_D * (float)j);
        float sn, cn;
        __sincosf(pos * inv, &sn, &cn);
        float* tb = (float*)table4;
        const int q = p * (DHEAD / 4) + (j >> 1);   // which float4
        tb[q * 4 + (j & 1) * 2 + 0] = cn;
        tb[q * 4 + (j & 1) * 2 + 1] = sn;
    }

    if (tid < 32) {
        __builtin_amdgcn_s_wait_tensorcnt(0);       // tile resident in LDS
    }
    __syncthreads();                                // all waves see tile + table

    // ---- Rotate: 2048 float4 / 256 threads = 8 iterations.
    //      idx = tid + 256*i: f = idx&127 = tid&127 is loop-invariant
    //      (table read hoisted); rowLocal = (tid>>7) + 2*i walks the rows.
    #pragma unroll
    for (int i = 0; i < TILE_V4 / 256; ++i) {
        const int idx      = tid + i * 256;
        const int rowLocal = idx >> 7;
        const int f        = idx & 127;
        const int posLocal = f >> 4;
        const int t        = f & 15;

        const v4f v  = tile4[idx];                       // ds_load_b128
        const v4f cs = table4[posLocal * 16 + t];        // hoisted ds_load_b128

        v4f o;
        o.x = v.x * cs.x - v.y * cs.y;   // xe*cos - xo*sin
        o.y = v.x * cs.y + v.y * cs.x;   // xe*sin + xo*cos
        o.z = v.z * cs.z - v.w * cs.w;
        o.w = v.z * cs.w + v.w * cs.z;

        const size_t goff = gbase + (size_t)rowLocal * rowStride + (size_t)f * 4;
        // Streaming output, zero reuse: non-temporal b128 store.
        __builtin_nontemporal_store(o, (v4f*)(out + goff));
    }
}

extern "C" void kernel_launch(void* const* d_in, const int* in_sizes, int n_in,
                              void* d_out, int out_size, void* d_ws, size_t ws_size,
                              hipStream_t stream) {
    const float* x  = (const float*)d_in[0];
    const int*   tp = (const int*)d_in[1];
    float*       out = (float*)d_out;

    const int total = in_sizes[0];          // B*H*S*D
    const int S     = in_sizes[1];          // 8192
    const int R     = total / (S * DHEAD);  // B*H = 128

    dim3 grid(S / TILE_S, R / TILE_R);      // 1024 x 8 = 8192 blocks
    rope_tdm_kernel<<<grid, 256, 0, stream>>>(x, tp, out, S);
}